// GATMinGRU_20332375179738
// MI455X (gfx1250) — compile-verified
//
#include <hip/hip_runtime.h>
#include <math.h>

#define NNODES 100000
#define NEDGES 1600000
#define IND 128
#define HD 64
#define NEG_SLOPE 0.2f

typedef float v2f __attribute__((ext_vector_type(2)));
typedef float v8f __attribute__((ext_vector_type(8)));

// f32 WMMA: D(16x16,f32) = A(16x4,f32) x B(4x16,f32) + C
__device__ __forceinline__ v8f wmma4(v2f a, v2f b, v8f c) {
    return __builtin_amdgcn_wmma_f32_16x16x4_f32(false, a, false, b, (short)0, c,
                                                 false, false);
}

// monotonic float <-> ordered-uint mapping for atomicMax-based segment max
__device__ __forceinline__ unsigned f2o(float f) {
    unsigned u = __float_as_uint(f);
    return (u & 0x80000000u) ? ~u : (u | 0x80000000u);
}
__device__ __forceinline__ float o2f(unsigned e) {
    unsigned u = (e & 0x80000000u) ? (e & 0x7FFFFFFFu) : ~e;
    return __uint_as_float(u);
}

__device__ __forceinline__ float lrelu(float v) {
    return v > 0.0f ? v : NEG_SLOPE * v;
}

// ---------------------------------------------------------------- init
__global__ void k_init(float* agg, float* denom, unsigned* maxe, int n64, int n) {
    int i = blockIdx.x * blockDim.x + threadIdx.x;
    if (i < n64) agg[i] = 0.0f;
    if (i < n) {
        denom[i] = 0.0f;
        maxe[i]  = 0x007FFFFFu;  // f2o(-inf)
    }
}

// -------------------------------------------- h = x @ W_gat^T  [N,128]x[128,64]
__global__ void k_gemm_h(const float* __restrict__ x, const float* __restrict__ Wg,
                         float* __restrict__ hbuf, int nTiles) {
    int wid = blockIdx.x * (blockDim.x >> 5) + (threadIdx.x >> 5);
    if (wid >= nTiles) return;                 // wave-uniform exit (EXEC stays full)
    int lane = threadIdx.x & 31;
    int mi = lane & 15, kh = lane >> 4;
    int row0 = wid * 16;

    v8f acc[4] = {};
    for (int k0 = 0; k0 < IND; k0 += 4) {
        int ka = k0 + 2 * kh;
        v2f a;
        a.x = x[(size_t)(row0 + mi) * IND + ka];
        a.y = x[(size_t)(row0 + mi) * IND + ka + 1];
#pragma unroll
        for (int t = 0; t < 4; t++) {
            int nrow = t * 16 + mi;            // B[k][n] = Wg[n][k]
            v2f b;
            b.x = Wg[(size_t)nrow * IND + ka];
            b.y = Wg[(size_t)nrow * IND + ka + 1];
            acc[t] = wmma4(a, b, acc[t]);
        }
    }
    int mb = (lane < 16) ? 0 : 8;
#pragma unroll
    for (int t = 0; t < 4; t++)
#pragma unroll
        for (int r = 0; r < 8; r++)
            hbuf[(size_t)(row0 + r + mb) * HD + t * 16 + mi] = acc[t][r];
}

// ---------------------------------------------------- a1 = h.attn_l, a2 = h.attn_r
__global__ void k_att(const float* __restrict__ h, const float* __restrict__ al,
                      const float* __restrict__ ar, float* __restrict__ a1,
                      float* __restrict__ a2, int n) {
    int i = blockIdx.x * blockDim.x + threadIdx.x;
    if (i >= n) return;
    float s1 = 0.0f, s2 = 0.0f;
    for (int k = 0; k < HD; k++) {
        float v = h[(size_t)i * HD + k];
        s1 += v * al[k];
        s2 += v * ar[k];
    }
    a1[i] = s1;
    a2[i] = s2;
}

// ---------------------------------------------------------------- segment max
__global__ void k_edge_max(const int* __restrict__ src, const int* __restrict__ dst,
                           const float* __restrict__ a1, const float* __restrict__ a2,
                           unsigned* __restrict__ maxe, int ne) {
    int e = blockIdx.x * blockDim.x + threadIdx.x;
    if (e >= ne) return;
    int s = src[e], d = dst[e];
    float v = lrelu(a1[s] + a2[d]);
    atomicMax(&maxe[d], f2o(v));
}

// ------------------------------------------------------- segment sum of exp(e-m)
__global__ void k_edge_sum(const int* __restrict__ src, const int* __restrict__ dst,
                           const float* __restrict__ a1, const float* __restrict__ a2,
                           const unsigned* __restrict__ maxe, float* __restrict__ denom,
                           int ne) {
    int e = blockIdx.x * blockDim.x + threadIdx.x;
    if (e >= ne) return;
    int s = src[e], d = dst[e];
    float v = lrelu(a1[s] + a2[d]);
    atomicAdd(&denom[d], __expf(v - o2f(maxe[d])));
}

// ------------------------------------ agg[dst] += alpha * h[src]  (wave per edge)
__global__ void k_edge_msg(const int* __restrict__ src, const int* __restrict__ dst,
                           const float* __restrict__ a1, const float* __restrict__ a2,
                           const unsigned* __restrict__ maxe,
                           const float* __restrict__ denom,
                           const float* __restrict__ h, float* __restrict__ agg,
                           int ne) {
    int wid = blockIdx.x * (blockDim.x >> 5) + (threadIdx.x >> 5);
    if (wid >= ne) return;
    int lane = threadIdx.x & 31;
    int s = src[wid], d = dst[wid];
    float v = lrelu(a1[s] + a2[d]);
    float alpha = __expf(v - o2f(maxe[d])) / denom[d];
    atomicAdd(&agg[(size_t)d * HD + lane],      h[(size_t)s * HD + lane] * alpha);
    atomicAdd(&agg[(size_t)d * HD + lane + 32], h[(size_t)s * HD + lane + 32] * alpha);
}

// ------------- MinGRU: out = (1-z)*hprev + z*tanh(x@Wh^T+bh), z=sig(x@Wz^T+bz)
__global__ void k_gru(const float* __restrict__ xin, const float* __restrict__ xbias,
                      const float* __restrict__ Wz, const float* __restrict__ bz,
                      const float* __restrict__ Wh, const float* __restrict__ bh,
                      const float* __restrict__ hprev, float* __restrict__ outp,
                      int nTiles) {
    int wid = blockIdx.x * (blockDim.x >> 5) + (threadIdx.x >> 5);
    if (wid >= nTiles) return;
    int lane = threadIdx.x & 31;
    int mi = lane & 15, kh = lane >> 4;
    int row0 = wid * 16;

    v8f az[4] = {};
    v8f ah[4] = {};
    for (int k0 = 0; k0 < HD; k0 += 4) {
        int ka = k0 + 2 * kh;
        v2f a;
        a.x = xin[(size_t)(row0 + mi) * HD + ka];
        a.y = xin[(size_t)(row0 + mi) * HD + ka + 1];
        if (xbias) {                            // kernel-arg uniform branch
            a.x += xbias[ka];
            a.y += xbias[ka + 1];
        }
#pragma unroll
        for (int t = 0; t < 4; t++) {
            int nrow = t * 16 + mi;
            v2f bzv, bhv;
            bzv.x = Wz[(size_t)nrow * HD + ka];
            bzv.y = Wz[(size_t)nrow * HD + ka + 1];
            bhv.x = Wh[(size_t)nrow * HD + ka];
            bhv.y = Wh[(size_t)nrow * HD + ka + 1];
            az[t] = wmma4(a, bzv, az[t]);
            ah[t] = wmma4(a, bhv, ah[t]);
        }
    }
    int mb = (lane < 16) ? 0 : 8;
#pragma unroll
    for (int t = 0; t < 4; t++) {
        int col = t * 16 + mi;
        float bzc = bz[col], bhc = bh[col];
#pragma unroll
        for (int r = 0; r < 8; r++) {
            int row = row0 + r + mb;
            float z  = 1.0f / (1.0f + __expf(-(az[t][r] + bzc)));
            float ht = tanhf(ah[t][r] + bhc);
            float hp = hprev[(size_t)row * HD + col];
            outp[(size_t)row * HD + col] = (1.0f - z) * hp + z * ht;
        }
    }
}

// ------------------------- event heads (3x 64->16 WMMA GEMMs) + time prediction
__global__ void k_heads(const float* __restrict__ h2,
                        const float* __restrict__ We1, const float* __restrict__ be1,
                        const float* __restrict__ We2, const float* __restrict__ be2,
                        const float* __restrict__ We3, const float* __restrict__ be3,
                        const float* __restrict__ Wfc, const float* __restrict__ bfc,
                        float* __restrict__ dout, int nTiles) {
    int wid = blockIdx.x * (blockDim.x >> 5) + (threadIdx.x >> 5);
    if (wid >= nTiles) return;
    int lane = threadIdx.x & 31;
    int mi = lane & 15, kh = lane >> 4;
    int row0 = wid * 16;

    v8f c1 = {}, c2 = {}, c3 = {};
    for (int k0 = 0; k0 < HD; k0 += 4) {
        int ka = k0 + 2 * kh;
        v2f a;
        a.x = h2[(size_t)(row0 + mi) * HD + ka];
        a.y = h2[(size_t)(row0 + mi) * HD + ka + 1];
        v2f b1, b2, b3;
        b1.x = We1[(size_t)mi * HD + ka]; b1.y = We1[(size_t)mi * HD + ka + 1];
        b2.x = We2[(size_t)mi * HD + ka]; b2.y = We2[(size_t)mi * HD + ka + 1];
        b3.x = We3[(size_t)mi * HD + ka]; b3.y = We3[(size_t)mi * HD + ka + 1];
        c1 = wmma4(a, b1, c1);
        c2 = wmma4(a, b2, c2);
        c3 = wmma4(a, b3, c3);
    }
    int mb = (lane < 16) ? 0 : 8;
#pragma unroll
    for (int r = 0; r < 8; r++) {
        int row = row0 + r + mb;
        dout[(size_t)row * 48 + mi]      = c1[r] + be1[mi];
        dout[(size_t)row * 48 + 16 + mi] = c2[r] + be2[mi];
        dout[(size_t)row * 48 + 32 + mi] = c3[r] + be3[mi];
    }
    // time head: lanes 0..15 each handle one node (after all WMMAs -> safe divergence)
    if (lane < 16) {
        int row = row0 + lane;
        float s = 0.0f;
        for (int k = 0; k < HD; k++) s += h2[(size_t)row * HD + k] * Wfc[k];
        dout[(size_t)NNODES * 48 + row] = s + bfc[0];
    }
}

extern "C" void kernel_launch(void* const* d_in, const int* in_sizes, int n_in,
                              void* d_out, int out_size, void* d_ws, size_t ws_size,
                              hipStream_t stream) {
    (void)in_sizes; (void)n_in; (void)out_size; (void)ws_size;

    const float* x   = (const float*)d_in[0];
    const int*   ei  = (const int*)d_in[1];
    const float* hp1 = (const float*)d_in[2];
    const float* hp2 = (const float*)d_in[3];
    const float* Wg  = (const float*)d_in[4];
    const float* al  = (const float*)d_in[5];
    const float* ar  = (const float*)d_in[6];
    const float* bg  = (const float*)d_in[7];
    const float* Wz1 = (const float*)d_in[8];
    const float* bz1 = (const float*)d_in[9];
    const float* Wh1 = (const float*)d_in[10];
    const float* bh1 = (const float*)d_in[11];
    const float* Wz2 = (const float*)d_in[12];
    const float* bz2 = (const float*)d_in[13];
    const float* Wh2 = (const float*)d_in[14];
    const float* bh2 = (const float*)d_in[15];
    const float* We1 = (const float*)d_in[16];
    const float* be1 = (const float*)d_in[17];
    const float* We2 = (const float*)d_in[18];
    const float* be2 = (const float*)d_in[19];
    const float* We3 = (const float*)d_in[20];
    const float* be3 = (const float*)d_in[21];
    const float* Wfc = (const float*)d_in[22];
    const float* bfc = (const float*)d_in[23];
    float* dout = (float*)d_out;

    const int* srcI = ei;
    const int* dstI = ei + NEDGES;

    // workspace layout (floats): hbuf[N*64] | gbuf[N*64] | a1[N] | a2[N] | denom[N] | maxe[N]
    float* ws   = (float*)d_ws;
    float* hbuf = ws;                                // h, later reused for h1
    float* gbuf = hbuf + (size_t)NNODES * HD;        // agg, later reused for h2
    float* a1   = gbuf + (size_t)NNODES * HD;
    float* a2   = a1 + NNODES;
    float* denom = a2 + NNODES;
    unsigned* maxe = (unsigned*)(denom + NNODES);

    const int tiles = NNODES / 16;                   // 6250 exactly
    const int tb = (tiles + 3) / 4;                  // 4 waves / 128-thread block

    int n64 = NNODES * HD;
    k_init<<<(n64 + 255) / 256, 256, 0, stream>>>(gbuf, denom, maxe, n64, NNODES);
    k_gemm_h<<<tb, 128, 0, stream>>>(x, Wg, hbuf, tiles);
    k_att<<<(NNODES + 127) / 128, 128, 0, stream>>>(hbuf, al, ar, a1, a2, NNODES);
    k_edge_max<<<(NEDGES + 255) / 256, 256, 0, stream>>>(srcI, dstI, a1, a2, maxe, NEDGES);
    k_edge_sum<<<(NEDGES + 255) / 256, 256, 0, stream>>>(srcI, dstI, a1, a2, maxe, denom, NEDGES);
    k_edge_msg<<<(NEDGES + 7) / 8, 256, 0, stream>>>(srcI, dstI, a1, a2, maxe, denom,
                                                     hbuf, gbuf, NEDGES);
    // MinGRU 1: in = agg + b_gat, state hp1 -> h1 (overwrites hbuf)
    k_gru<<<tb, 128, 0, stream>>>(gbuf, bg, Wz1, bz1, Wh1, bh1, hp1, hbuf, tiles);
    // MinGRU 2: in = h1, state hp2 -> h2 (overwrites gbuf)
    k_gru<<<tb, 128, 0, stream>>>(hbuf, nullptr, Wz2, bz2, Wh2, bh2, hp2, gbuf, tiles);
    k_heads<<<tb, 128, 0, stream>>>(gbuf, We1, be1, We2, be2, We3, be3, Wfc, bfc,
                                    dout, tiles);
}